// SAGEConv_4964982194658
// MI455X (gfx1250) — compile-verified
//
#include <hip/hip_runtime.h>

#define N_NODES 40000
#define N_EDGES 640000
#define D 128
#define LDSP 132  // padded LDS row stride (floats) to avoid 16-way bank conflicts

typedef __attribute__((ext_vector_type(2))) float v2f;
typedef __attribute__((ext_vector_type(8))) float v8f;

// ---------------- Kernel 1: zero the aggregation buffers ----------------
__global__ __launch_bounds__(256) void sage_zero(float* __restrict__ sum,
                                                 float* __restrict__ deg) {
    int i = blockIdx.x * 256 + threadIdx.x;
    if (i < N_NODES * D) {
        sum[i] = 0.0f;
    } else if (i < N_NODES * D + N_NODES) {
        deg[i - N_NODES * D] = 0.0f;
    }
}

// ---------------- Kernel 2: edge scatter (gather src feat, atomic-add into dst) ----
// 32 lanes per edge, float4 per lane -> 128 floats per edge. feat + sum are
// L2-resident (20.5 MB each on 192 MB L2), so these atomics run at L2 rate.
__global__ __launch_bounds__(256) void sage_scatter(const float* __restrict__ feat,
                                                    const int* __restrict__ src,
                                                    const int* __restrict__ dst,
                                                    float* __restrict__ sum,
                                                    float* __restrict__ deg) {
    unsigned gtid = blockIdx.x * 256u + threadIdx.x;
    unsigned e = gtid >> 5;
    unsigned c = gtid & 31u;
    if (e >= N_EDGES) return;
    int s = src[e];
    int d = dst[e];
    const float4 v = *(const float4*)(feat + (size_t)s * D + c * 4);
    float* p = sum + (size_t)d * D + c * 4;
    __hip_atomic_fetch_add(p + 0, v.x, __ATOMIC_RELAXED, __HIP_MEMORY_SCOPE_AGENT);
    __hip_atomic_fetch_add(p + 1, v.y, __ATOMIC_RELAXED, __HIP_MEMORY_SCOPE_AGENT);
    __hip_atomic_fetch_add(p + 2, v.z, __ATOMIC_RELAXED, __HIP_MEMORY_SCOPE_AGENT);
    __hip_atomic_fetch_add(p + 3, v.w, __ATOMIC_RELAXED, __HIP_MEMORY_SCOPE_AGENT);
    if (c == 0) {
        __hip_atomic_fetch_add(deg + d, 1.0f, __ATOMIC_RELAXED, __HIP_MEMORY_SCOPE_AGENT);
    }
}

// ---------------- Kernel 3: dual GEMM + bias via V_WMMA_F32_16X16X4_F32 ----------
// Block = 256 threads (8 waves), covers 64 nodes (4 M-tiles of 16).
// Tiles staged with GLOBAL_LOAD_ASYNC_TO_LDS_B128 (ASYNCcnt). The degree
// division is folded into the epilogue: out = accS + inv[m] * accN + bias,
// identical to (sum*inv) @ Wn since the scaling is row-wise.
__global__ __launch_bounds__(256) void sage_gemm(const float* __restrict__ feat,
                                                 const float* __restrict__ deg,
                                                 const float* __restrict__ Ws,
                                                 const float* __restrict__ bs,
                                                 const float* __restrict__ Wn,
                                                 const float* __restrict__ bn,
                                                 float* __restrict__ out) {
    __shared__ float sF[64 * LDSP];   // feat tile
    __shared__ float sH[64 * LDSP];   // raw neighbor-sum tile
    __shared__ float sInv[64];

    const int t = threadIdx.x;
    const int node0 = blockIdx.x * 64;

    if (t < 64) {
        sInv[t] = 1.0f / fmaxf(deg[node0 + t], 1.0f);
    }

    // Async-stage tiles: 64x128 floats = 2048 b128 transfers per buffer,
    // 8 per thread per buffer. LDS dest address = low 32 bits of generic ptr.
    const unsigned baseF = (unsigned)(uintptr_t)&sF[0];
    const unsigned baseH = (unsigned)(uintptr_t)&sH[0];
#pragma unroll
    for (int i = 0; i < 8; ++i) {
        int idx = t + i * 256;      // 0..2047
        int r = idx >> 5;
        int c4 = idx & 31;
        unsigned lb = (unsigned)((r * LDSP + c4 * 4) * 4);   // byte offset in LDS
        unsigned long long gaF =
            (unsigned long long)(feat + (size_t)(node0 + r) * D + c4 * 4);
        unsigned long long gaH =
            (unsigned long long)(out + (size_t)(node0 + r) * D + c4 * 4);
        asm volatile("global_load_async_to_lds_b128 %0, %1, off"
                     :: "v"(baseF + lb), "v"(gaF) : "memory");
        asm volatile("global_load_async_to_lds_b128 %0, %1, off"
                     :: "v"(baseH + lb), "v"(gaH) : "memory");
    }
    asm volatile("s_wait_asynccnt 0x0" ::: "memory");
    __syncthreads();

    const int wid = t >> 5;
    const int lane = t & 31;
    const int lh = lane & 15;
    const int khi = (lane >> 4) << 1;   // lanes 16..31 hold K+2, K+3
    const int ncol = wid * 16 + lh;

    v8f accS[4] = {v8f{}, v8f{}, v8f{}, v8f{}};
    v8f accN[4] = {v8f{}, v8f{}, v8f{}, v8f{}};

    for (int k0 = 0; k0 < D; k0 += 4) {
        const int k = k0 + khi;
        // B fragments (4x16): lane<16 -> rows k0,k0+1; lane>=16 -> rows k0+2,k0+3
        v2f bS, bN;
        bS.x = Ws[k * D + ncol];
        bS.y = Ws[(k + 1) * D + ncol];
        bN.x = Wn[k * D + ncol];
        bN.y = Wn[(k + 1) * D + ncol];
#pragma unroll
        for (int mt = 0; mt < 4; ++mt) {
            const int row = mt * 16 + lh;
            // A fragments (16x4) from LDS (8-byte aligned: k even, LDSP even)
            v2f aS = *(const v2f*)&sF[row * LDSP + k];
            v2f aN = *(const v2f*)&sH[row * LDSP + k];
            accS[mt] = __builtin_amdgcn_wmma_f32_16x16x4_f32(
                false, aS, false, bS, (short)0, accS[mt], false, false);
            accN[mt] = __builtin_amdgcn_wmma_f32_16x16x4_f32(
                false, aN, false, bN, (short)0, accN[mt], false, false);
        }
    }

    // Epilogue: D layout — VGPR r: lanes 0-15 M=r, lanes 16-31 M=r+8, N=ncol
    const float bias = bs[ncol] + bn[ncol];
    const int mbase = (lane >> 4) ? 8 : 0;
#pragma unroll
    for (int mt = 0; mt < 4; ++mt) {
#pragma unroll
        for (int r = 0; r < 8; ++r) {
            int mrow = mt * 16 + mbase + r;
            float v = fmaf(accN[mt][r], sInv[mrow], accS[mt][r]) + bias;
            out[(size_t)(node0 + mrow) * D + ncol] = v;
        }
    }
}

extern "C" void kernel_launch(void* const* d_in, const int* in_sizes, int n_in,
                              void* d_out, int out_size, void* d_ws, size_t ws_size,
                              hipStream_t stream) {
    const float* feat = (const float*)d_in[0];
    const int*   src  = (const int*)d_in[1];
    const int*   dst  = (const int*)d_in[2];
    const float* Ws   = (const float*)d_in[3];
    const float* bs   = (const float*)d_in[4];
    const float* Wn   = (const float*)d_in[5];
    const float* bn   = (const float*)d_in[6];
    float* out = (float*)d_out;
    float* deg = (float*)d_ws;   // 40000 floats of scratch

    // 1) zero sum (in d_out, used as scratch accumulator) and degree (in d_ws)
    {
        int total = N_NODES * D + N_NODES;
        sage_zero<<<(total + 255) / 256, 256, 0, stream>>>(out, deg);
    }
    // 2) edge scatter-add: 32 lanes per edge
    {
        int blocks = (N_EDGES * 32) / 256;   // 80000
        sage_scatter<<<blocks, 256, 0, stream>>>(feat, src, dst, out, deg);
    }
    // 3) mean + dual GEMM + bias, 64 nodes per block (in-place on d_out)
    {
        sage_gemm<<<N_NODES / 64, 256, 0, stream>>>(feat, deg, Ws, bs, Wn, bn, out);
    }
}